// Dense_with_person_13580686590254
// MI455X (gfx1250) — compile-verified
//
#include <hip/hip_runtime.h>

typedef __attribute__((ext_vector_type(2))) float v2f;
typedef __attribute__((ext_vector_type(8))) float v8f;

#define LDS_ROW_F 36                 // 144-byte padded row stride -> conflict-free ds_load_b64
#define BUF_F     (16 * LDS_ROW_F)   // one (stage,input) buffer: 16 rows = 576 floats
#define WAVE_F    (4 * BUF_F)        // 2 stages x 2 inputs per wave = 2304 floats (9216 B)

// D = A(16x4) * B(4x16) + C via CDNA5 WMMA, f32 path.
__device__ __forceinline__ v8f wmma_acc(v2f a, v2f b, v8f c) {
    return __builtin_amdgcn_wmma_f32_16x16x4_f32(
        false, a, false, b, (short)0, c, false, false);
}

// Async copy, GVS mode: scalar 64-bit base + 32-bit per-lane offset.
// 32 lanes x 16B = 4 full rows (128B each), fully coalesced, tracked by ASYNCcnt.
__device__ __forceinline__ void async_copy16(unsigned ldsOff, unsigned voff,
                                             unsigned long long sbase) {
    asm volatile("global_load_async_to_lds_b128 %0, %1, %2"
                 :: "v"(ldsOff), "v"(voff), "s"(sbase)
                 : "memory");
}

template <int N>
__device__ __forceinline__ void wait_asynccnt() {
    asm volatile("s_wait_asynccnt %0" :: "n"(N) : "memory");
}

__device__ __forceinline__ void wait_dscnt0() {
    asm volatile("s_wait_dscnt 0x0" ::: "memory");
}

// All 16 columns of the accumulator hold the same row-sum (B was all-ones).
__device__ __forceinline__ float sel8(v8f c, int r) {
    float s = c[0];
    s = (r == 1) ? c[1] : s;
    s = (r == 2) ? c[2] : s;
    s = (r == 3) ? c[3] : s;
    s = (r == 4) ? c[4] : s;
    s = (r == 5) ? c[5] : s;
    s = (r == 6) ? c[6] : s;
    s = (r == 7) ? c[7] : s;
    return s;
}

__global__ __launch_bounds__(256) void rowcorr_wmma_async_kernel(
    const float* __restrict__ v1, const float* __restrict__ v2,
    float* __restrict__ out, int tiles) {
    constexpr int D = 256;
    __shared__ __align__(16) float smem[8 * WAVE_F];   // 73,728 B per block

    const int lane = threadIdx.x & 31;
    const int wave = threadIdx.x >> 5;
    const int tile = blockIdx.x * 8 + wave;            // one 16-row tile per wave
    if (tile >= tiles) return;                         // wave-uniform scalar branch

    const long rowBase = (long)tile * 16;

    // ---- staging-side lane mapping: 4 rows x 8 lanes x 16B per async instruction
    const int lr = lane >> 3;                          // row within 4-row group
    const int lc = lane & 7;                           // 16B slot within the 128B row chunk
    const unsigned voBase =
        (unsigned)(((rowBase + lr) * D + lc * 4) * 4); // per-lane global byte offset
    const unsigned long long b1 = (unsigned long long)v1;
    const unsigned long long b2 = (unsigned long long)v2;
    const int ldsWave = wave * WAVE_F;

    // ---- WMMA A-layout lane mapping (lane holds row m, K-pair selected by half)
    const int m = lane & 15;
    const int h = lane >> 4;

    const v2f ones = {1.0f, 1.0f};                     // B = 4x16 ones -> D accumulates row sums
    v8f cx = {}, cy = {}, cxy = {}, cxx = {}, cyy = {};

    auto issue_stage = [&](int s) {
        wait_dscnt0();                                 // WAR guard: prior reads of this buffer drained
        const int p = s & 1;
        const unsigned co = (unsigned)(s * 32 * 4);    // 32 columns = 128 B per stage
#pragma unroll
        for (int i = 0; i < 4; ++i) {
            const int ro = (4 * i + lr) * LDS_ROW_F + lc * 4;
            unsigned o1 = (unsigned)(uintptr_t)&smem[ldsWave + (p * 2 + 0) * BUF_F + ro];
            unsigned o2 = (unsigned)(uintptr_t)&smem[ldsWave + (p * 2 + 1) * BUF_F + ro];
            const unsigned go = voBase + co + (unsigned)(i * 4 * D * 4);
            async_copy16(o1, go, b1);
            async_copy16(o2, go, b2);
        }
    };

    auto consume_stage = [&](int s) {
        const int p = s & 1;
        const int bx = ldsWave + (p * 2 + 0) * BUF_F + m * LDS_ROW_F + 2 * h;
        const int by = ldsWave + (p * 2 + 1) * BUF_F + m * LDS_ROW_F + 2 * h;
#pragma unroll
        for (int k = 0; k < 32; k += 4) {              // 8 WMMA k-chunks per stage
            v2f x = *(const v2f*)&smem[bx + k];        // ds_load_b64, conflict-free banks
            v2f y = *(const v2f*)&smem[by + k];
            v2f xy = x * y;
            v2f xx = x * x;
            v2f yy = y * y;
            cx  = wmma_acc(x,  ones, cx);
            cy  = wmma_acc(y,  ones, cy);
            cxy = wmma_acc(xy, ones, cxy);
            cxx = wmma_acc(xx, ones, cxx);
            cyy = wmma_acc(yy, ones, cyy);
        }
    };

    issue_stage(0);
#pragma unroll
    for (int s = 0; s < 8; ++s) {
        if (s < 7) {
            issue_stage(s + 1);
            wait_asynccnt<8>();                        // in-order: stage s's 8 copies landed
        } else {
            wait_asynccnt<0>();
        }
        consume_stage(s);
    }

    // ---- epilogue: row r (0..7) = component r lanes 0-15; row r+8 = component r lanes 16-31
    const int r = lane & 7;
    const float sx  = sel8(cx,  r);
    const float sy  = sel8(cy,  r);
    const float sxy = sel8(cxy, r);
    const float sxx = sel8(cxx, r);
    const float syy = sel8(cyy, r);

    const bool writer = (lane < 8) | (lane >= 16 && lane < 24);
    if (writer) {
        const int row = r + ((lane >= 16) ? 8 : 0);
        const float inv_d = 1.0f / (float)D;
        const float ex  = sx * inv_d;
        const float ey  = sy * inv_d;
        const float cov = sxy * inv_d - ex * ey;       // Exy - Ex*Ey
        const float var1 = (sxx - (float)D * ex * ex) * (1.0f / (float)(D - 1));
        const float var2 = (syy - (float)D * ey * ey) * (1.0f / (float)(D - 1));
        out[rowBase + row] = cov * __frsqrt_rn(var1 * var2);
    }
}

extern "C" void kernel_launch(void* const* d_in, const int* in_sizes, int n_in,
                              void* d_out, int out_size, void* d_ws, size_t ws_size,
                              hipStream_t stream) {
    const float* v1 = (const float*)d_in[0];
    const float* v2 = (const float*)d_in[1];
    float* out = (float*)d_out;

    const int D = 256;
    const int Nrows = in_sizes[0] / D;                 // 262144
    const int tiles = Nrows / 16;                      // 16384
    const int blocks = (tiles + 7) / 8;                // 8 waves per 256-thread block

    rowcorr_wmma_async_kernel<<<blocks, 256, 0, stream>>>(v1, v2, out, tiles);
}